// DSTAGNN_block_39333310497321
// MI455X (gfx1250) — compile-verified
//
#include <hip/hip_runtime.h>
#include <math.h>

#define Bq 32
#define Nn 716
#define Tt 12
#define DM 512
#define HT 3
#define KC 3
#define CO 64
#define SCALE 0.17677669529663687f  // 1/sqrt(32)

typedef __attribute__((ext_vector_type(2))) float v2f;
typedef __attribute__((ext_vector_type(8))) float v8f;

__device__ __forceinline__ float wsum(float v){
#pragma unroll
  for(int o=16;o>0;o>>=1) v += __shfl_xor(v,o,32);
  return v;
}
__device__ __forceinline__ float wmaxr(float v){
#pragma unroll
  for(int o=16;o>0;o>>=1) v = fmaxf(v,__shfl_xor(v,o,32));
  return v;
}

// ---------------- K1: temporal embedding + LN over N --------------------
__global__ void k_temb(const float* __restrict__ x, const float* __restrict__ posT,
                       const float* __restrict__ w, const float* __restrict__ bvec,
                       float* __restrict__ TE){
  int row = blockIdx.x; int bb = row/Tt, t = row%Tt;
  __shared__ float r0[8], r1[8];
  float s=0.f, s2=0.f;
  for(int n=threadIdx.x;n<Nn;n+=256){
    float v = x[(bb*Nn+n)*Tt+t] + posT[t*Nn+n];
    s+=v; s2+=v*v;
  }
  s=wsum(s); s2=wsum(s2);
  if((threadIdx.x&31)==0){ r0[threadIdx.x>>5]=s; r1[threadIdx.x>>5]=s2; }
  __syncthreads();
  float ts=0.f, ts2=0.f;
  for(int i=0;i<8;i++){ ts+=r0[i]; ts2+=r1[i]; }
  float mean=ts/(float)Nn, var=ts2/(float)Nn-mean*mean, rs=rsqrtf(var+1e-5f);
  for(int n=threadIdx.x;n<Nn;n+=256){
    float v = x[(bb*Nn+n)*Tt+t] + posT[t*Nn+n];
    TE[row*Nn+n] = (v-mean)*rs*w[n]+bvec[n];
  }
}

// ---------------- K2: temporal q,k,v projections ------------------------
__global__ void k_tqkv(const float* __restrict__ TE, const float* __restrict__ Wq,
                       const float* __restrict__ Wk, const float* __restrict__ Wv,
                       float* __restrict__ q, float* __restrict__ k, float* __restrict__ v){
  __shared__ float rowv[Nn];
  int row = blockIdx.x;
  for(int n=threadIdx.x;n<Nn;n+=96) rowv[n]=TE[row*Nn+n];
  __syncthreads();
  int j=threadIdx.x;
  float aq=0.f, ak=0.f, av=0.f;
  for(int n=0;n<Nn;n++){
    float t=rowv[n];
    aq+=t*Wq[n*96+j]; ak+=t*Wk[n*96+j]; av+=t*Wv[n*96+j];
  }
  q[row*96+j]=aq; k[row*96+j]=ak; v[row*96+j]=av;
}

// ---------------- K3: temporal attention (softmax over t-axis!) ---------
__global__ void k_tattn(const float* __restrict__ q, const float* __restrict__ k,
                        const float* __restrict__ v, const float* __restrict__ resA,
                        float* __restrict__ scores_out, float* __restrict__ ctx){
  int bh=blockIdx.x; int bb=bh/HT, h=bh%HT;
  __shared__ float qs[12][32], ksm[12][32], vs[12][32], sc[12][12], at[12][12];
  int tid=threadIdx.x;
  if(tid<384){ int t=tid>>5, d=tid&31;
    qs[t][d]=q[(bb*Tt+t)*96 + h*32+d];
    ksm[t][d]=k[(bb*Tt+t)*96 + h*32+d];
    vs[t][d]=v[(bb*Tt+t)*96 + h*32+d];
  }
  __syncthreads();
  if(tid<144){ int t=tid/12, s=tid%12;
    float a=0.f;
    for(int d=0;d<32;d++) a+=qs[t][d]*ksm[s][d];
    a = a*SCALE + resA[((bb*HT+h)*Tt+t)*Tt+s];
    sc[t][s]=a;
    scores_out[((bb*HT+h)*Tt+t)*Tt+s]=a;   // second output of the tuple
  }
  __syncthreads();
  if(tid<12){ int s=tid;
    float mx=-1e30f;
    for(int t=0;t<12;t++) mx=fmaxf(mx, sc[t][s]);
    float sum=0.f;
    for(int t=0;t<12;t++){ float e=__expf(sc[t][s]-mx); at[t][s]=e; sum+=e; }
    float inv=1.f/sum;
    for(int t=0;t<12;t++) at[t][s]*=inv;
  }
  __syncthreads();
  if(tid<384){ int t=tid>>5, d=tid&31;
    float a=0.f;
    for(int s=0;s<12;s++) a+=at[t][s]*vs[s][d];
    ctx[(bb*Tt+t)*96 + h*32+d]=a;
  }
}

// ---------------- K4: ctx @ fc_t + TE, plain LN -------------------------
__global__ void k_tat(const float* __restrict__ ctx, const float* __restrict__ fc,
                      const float* __restrict__ TE, float* __restrict__ TAT){
  int row=blockIdx.x;
  __shared__ float c[96]; __shared__ float r0[8], r1[8];
  if(threadIdx.x<96) c[threadIdx.x]=ctx[row*96+threadIdx.x];
  __syncthreads();
  float s=0.f, s2=0.f;
  for(int n=threadIdx.x;n<Nn;n+=256){
    float a=TE[row*Nn+n];
    for(int j=0;j<96;j++) a+=c[j]*fc[j*Nn+n];
    TAT[row*Nn+n]=a; s+=a; s2+=a*a;
  }
  s=wsum(s); s2=wsum(s2);
  if((threadIdx.x&31)==0){ r0[threadIdx.x>>5]=s; r1[threadIdx.x>>5]=s2; }
  __syncthreads();
  float ts=0.f, ts2=0.f;
  for(int i=0;i<8;i++){ ts+=r0[i]; ts2+=r1[i]; }
  float mean=ts/(float)Nn, var=ts2/(float)Nn-mean*mean, rs=rsqrtf(var+1e-5f);
  for(int n=threadIdx.x;n<Nn;n+=256)
    TAT[row*Nn+n]=(TAT[row*Nn+n]-mean)*rs;
}

// ---------------- K5: pre_conv + pos_S + spatial LN ---------------------
__global__ void k_sememb(const float* __restrict__ TAT, const float* __restrict__ preW,
                         const float* __restrict__ preB, const float* __restrict__ posS,
                         const float* __restrict__ w, const float* __restrict__ bvec,
                         float* __restrict__ SE){
  int row=blockIdx.x; int b_=row/Nn, n=row%Nn;
  __shared__ float tt[12]; __shared__ float r0[8], r1[8];
  if(threadIdx.x<12) tt[threadIdx.x]=TAT[(b_*Tt+threadIdx.x)*Nn+n];
  __syncthreads();
  float vals[2], s=0.f, s2=0.f;
  for(int i=0;i<2;i++){
    int d=threadIdx.x+256*i;
    float a=preB[d]+posS[n*DM+d];
    for(int t=0;t<12;t++) a+=preW[d*Tt+t]*tt[t];
    vals[i]=a; s+=a; s2+=a*a;
  }
  s=wsum(s); s2=wsum(s2);
  if((threadIdx.x&31)==0){ r0[threadIdx.x>>5]=s; r1[threadIdx.x>>5]=s2; }
  __syncthreads();
  float ts=0.f, ts2=0.f;
  for(int i=0;i<8;i++){ ts+=r0[i]; ts2+=r1[i]; }
  float mean=ts/(float)DM, var=ts2/(float)DM-mean*mean, rs=rsqrtf(var+1e-5f);
  for(int i=0;i<2;i++){
    int d=threadIdx.x+256*i;
    SE[row*DM+d]=(vals[i]-mean)*rs*w[d]+bvec[d];
  }
}

// ---------------- K6: WMMA GEMM: SE(22912x512) @ [Wq_s|Wk_s](512x192) ---
__global__ void k_proj(const float* __restrict__ SE, const float* __restrict__ Wq,
                       const float* __restrict__ Wk, float* __restrict__ proj){
  int wave = threadIdx.x>>5, lane = threadIdx.x&31;
  int gt = blockIdx.x*8 + wave;
  int mt = gt/12, nt = gt%12;
  int m0 = mt*16, n0 = nt*16;
  const float* Wm = (nt<6) ? Wq : Wk;
  int c0 = (nt<6) ? n0 : (n0-96);
  int mrow = m0 + (lane&15);
  int koff = (lane<16) ? 0 : 2;
  int col  = c0 + (lane&15);
  v8f acc = {};
  for(int k0=0;k0<DM;k0+=4){
    v2f a;  a.x = SE[mrow*DM + k0+koff];      a.y = SE[mrow*DM + k0+koff+1];
    v2f bf; bf.x = Wm[(k0+koff)*96 + col];    bf.y = Wm[(k0+koff+1)*96 + col];
    acc = __builtin_amdgcn_wmma_f32_16x16x4_f32(false, a, false, bf, (short)0, acc, false, false);
  }
  int ocol = n0 + (lane&15);
  int rbase = m0 + ((lane<16)?0:8);
#pragma unroll
  for(int r=0;r<8;r++) proj[(rbase+r)*192 + ocol] = acc[r];
}

// ---------------- K7: fused spatial attn + cheb aggregation -------------
#define SROWS 720
__global__ void k_spat(const float* __restrict__ proj, const float* __restrict__ adj,
                       const float* __restrict__ cmask, const float* __restrict__ cpoly,
                       const float* __restrict__ x, float* __restrict__ rhs){
  int blk = blockIdx.x;
  int nt = blk % 45; int tmp = blk/45; int kk = tmp % KC; int b_ = tmp / KC;
  int n0 = nt*16;
  int tn = (Nn - n0 < 16) ? (Nn - n0) : 16;
  __shared__ float S[SROWS*16];
  __shared__ float Kt[16*32];
  int tid=threadIdx.x, lane=tid&31, wave=tid>>5;
  for(int i=tid;i<16*32;i+=256) Kt[i]=0.f;
  __syncthreads();
  for(int i=tid;i<tn*32;i+=256){
    int j=i>>5, d=i&31;
    Kt[j*32+d]=proj[(b_*Nn+n0+j)*192 + 96 + kk*32 + d];
  }
  __syncthreads();
  const float* qb = proj + (size_t)(b_*Nn)*192 + kk*32;
  int koff = (lane<16)?0:2;
  for(int mt=wave; mt<45; mt+=8){
    int m0=mt*16;
    int mrow = m0 + (lane&15);
    bool mval = (mrow < Nn);
    v8f acc = {};
    for(int k4=0;k4<32;k4+=4){
      v2f a;
      a.x = mval ? qb[mrow*192 + k4+koff]   : 0.f;
      a.y = mval ? qb[mrow*192 + k4+koff+1] : 0.f;
      v2f bf;
      bf.x = Kt[(lane&15)*32 + k4+koff];
      bf.y = Kt[(lane&15)*32 + k4+koff+1];
      acc = __builtin_amdgcn_wmma_f32_16x16x4_f32(false, a, false, bf, (short)0, acc, false, false);
    }
    int col = lane&15; int nglob = n0+col;
    int rb = (lane<16)?0:8;
#pragma unroll
    for(int r=0;r<8;r++){
      int m = m0 + rb + r;
      float sv = -1e30f;
      if(m < Nn && col < tn)
        sv = acc[r]*SCALE + adj[m*Nn+nglob]*cmask[((size_t)kk*Nn + m)*Nn + nglob];
      S[(m0+rb+r)*16 + col] = sv;
    }
  }
  __syncthreads();
  for(int j=wave; j<16; j+=8){
    if(j>=tn) continue;                 // uniform per wave
    int nglob=n0+j;
    float mx=-1e30f;
    for(int m=lane;m<Nn;m+=32) mx=fmaxf(mx,S[m*16+j]);
    mx = wmaxr(mx);
    float sum=0.f, acc[12];
#pragma unroll
    for(int t=0;t<12;t++) acc[t]=0.f;
    for(int m=lane;m<Nn;m+=32){
      float p=__expf(S[m*16+j]-mx);
      sum+=p;
      float cp = p * cpoly[((size_t)kk*Nn+m)*Nn+nglob];
      const float* xr = x + (b_*Nn+m)*Tt;
#pragma unroll
      for(int t=0;t<12;t++) acc[t]+=cp*xr[t];
    }
    sum = wsum(sum);
#pragma unroll
    for(int t=0;t<12;t++) acc[t]=wsum(acc[t]);
    if(lane==0){
      float inv=1.f/sum;
      float* o = rhs + ((b_*KC+kk)*Nn + nglob)*Tt;
#pragma unroll
      for(int t=0;t<12;t++) o[t]=acc[t]*inv;
    }
  }
}

// ---------------- K8: Theta projection + relu -> X (B,C,N,T) ------------
__global__ void k_theta(const float* __restrict__ rhs, const float* __restrict__ Th,
                        float* __restrict__ X){
  int i = blockIdx.x*256 + threadIdx.x;
  const int total = Bq*CO*Nn*Tt;
  if(i>=total) return;
  int t=i%Tt; int r=i/Tt; int n=r%Nn; r/=Nn; int o=r%CO; int b_=r/CO;
  float a=0.f;
#pragma unroll
  for(int k=0;k<KC;k++) a += rhs[((b_*KC+k)*Nn+n)*Tt+t]*Th[k*CO+o];
  X[i]=fmaxf(a,0.f);
}

// ---------------- K9: fused GTU(3/5/7) + fcmy + relu --------------------
// 8 (b,n) rows per 512-thread block: the 16 waves re-hit the same conv
// weights through WGP$/L2 -> ~8x less weight traffic than 1 row/block.
#define GTR 8
__global__ void k_gtu(const float* __restrict__ X,
                      const float* __restrict__ w3, const float* __restrict__ b3,
                      const float* __restrict__ w5, const float* __restrict__ b5,
                      const float* __restrict__ w7, const float* __restrict__ b7,
                      const float* __restrict__ fW, const float* __restrict__ fb,
                      float* __restrict__ tout){
  __shared__ float xs[GTR][CO][Tt];           // 24 KB
  int tid=threadIdx.x;                        // 512
  int c = tid & 63, rr = tid >> 6;            // rr in 0..7
  int r0 = blockIdx.x*GTR;
  for(int i=tid;i<GTR*CO*Tt;i+=512){
    int rrr=i/(CO*Tt); int rem=i%(CO*Tt); int cc=rem/Tt; int t2=rem%Tt;
    int row=r0+rrr; int b_=row/Nn, n=row%Nn;
    xs[rrr][cc][t2]=X[((b_*CO+cc)*Nn+n)*Tt+t2];
  }
  __syncthreads();
  float g[24]; int gi=0;
  { // kernel 3, 10 outputs
    const float* wA=w3+c*CO*3; const float* wB=w3+(c+CO)*CO*3;
    for(int s=0;s<10;s++){
      float a=b3[c], b2=b3[c+CO];
      for(int ci=0;ci<CO;ci++){
        const float* wa=wA+ci*3; const float* wb=wB+ci*3;
#pragma unroll
        for(int r=0;r<3;r++){ float xv=xs[rr][ci][s+r]; a+=xv*wa[r]; b2+=xv*wb[r]; }
      }
      g[gi++]=tanhf(a)*(1.f/(1.f+__expf(-b2)));
    }
  }
  { // kernel 5, 8 outputs
    const float* wA=w5+c*CO*5; const float* wB=w5+(c+CO)*CO*5;
    for(int s=0;s<8;s++){
      float a=b5[c], b2=b5[c+CO];
      for(int ci=0;ci<CO;ci++){
        const float* wa=wA+ci*5; const float* wb=wB+ci*5;
#pragma unroll
        for(int r=0;r<5;r++){ float xv=xs[rr][ci][s+r]; a+=xv*wa[r]; b2+=xv*wb[r]; }
      }
      g[gi++]=tanhf(a)*(1.f/(1.f+__expf(-b2)));
    }
  }
  { // kernel 7, 6 outputs
    const float* wA=w7+c*CO*7; const float* wB=w7+(c+CO)*CO*7;
    for(int s=0;s<6;s++){
      float a=b7[c], b2=b7[c+CO];
      for(int ci=0;ci<CO;ci++){
        const float* wa=wA+ci*7; const float* wb=wB+ci*7;
#pragma unroll
        for(int r=0;r<7;r++){ float xv=xs[rr][ci][s+r]; a+=xv*wa[r]; b2+=xv*wb[r]; }
      }
      g[gi++]=tanhf(a)*(1.f/(1.f+__expf(-b2)));
    }
  }
  int row=r0+rr; int b_=row/Nn, n=row%Nn;
#pragma unroll
  for(int t=0;t<Tt;t++){
    float a=fb[t];
#pragma unroll
    for(int s=0;s<24;s++) a+=g[s]*fW[s*Tt+t];
    tout[((b_*CO+c)*Nn+n)*Tt+t]=fmaxf(a,0.f);
  }
}

// ---------------- K10: one GRU step, LDS-tiled weights ------------------
// 16 rows per 256-thread block. Weights staged as 64x64 tiles (padded to
// 65 cols -> conflict-free LDS reads), loaded coalesced once per block:
// 16x less weight traffic than 1 row/block, and fully coalesced.
#define GR 16
__global__ void k_gru(const float* __restrict__ X, const float* __restrict__ Wih,
                      const float* __restrict__ Whh, const float* __restrict__ bih,
                      const float* __restrict__ bhh, float* __restrict__ h,
                      float* __restrict__ rout, int t){
  __shared__ float xs[GR][CO], hs[GR][CO];    // 8 KB
  __shared__ float wt[CO][CO+1];              // 16.25 KB, padded
  int tid=threadIdx.x;                        // 256
  int j = tid & 63;                           // output unit
  int sub = tid >> 6;                         // 0..3 -> rows sub*4..sub*4+3
  int r0 = blockIdx.x*GR;
  for(int i=tid;i<GR*CO;i+=256){
    int rr=i>>6, cc=i&63;
    int row=r0+rr; int b_=row/Nn, n=row%Nn;
    xs[rr][cc]=X[((b_*CO+cc)*Nn+n)*Tt+t];
    hs[rr][cc]=(t==0)?0.f:h[row*CO+cc];
  }
  float acc[6][4];
#pragma unroll
  for(int g6=0;g6<6;g6++)
#pragma unroll
    for(int r4=0;r4<4;r4++) acc[g6][r4]=0.f;
#pragma unroll
  for(int tile=0;tile<6;tile++){
    const float* Wsrc = (tile<3) ? (Wih + tile*CO*CO) : (Whh + (tile-3)*CO*CO);
    __syncthreads();                           // also covers xs/hs on tile 0
    for(int i=tid;i<CO*CO;i+=256) wt[i>>6][i&63]=Wsrc[i];
    __syncthreads();
#pragma unroll
    for(int r4=0;r4<4;r4++){
      int lr=sub*4+r4;
      const float* xv = (tile<3) ? xs[lr] : hs[lr];
      float a=0.f;
      for(int cc=0;cc<CO;cc++) a += xv[cc]*wt[j][cc];
      acc[tile][r4]=a;
    }
  }
#pragma unroll
  for(int r4=0;r4<4;r4++){
    int lr=sub*4+r4; int row=r0+lr; int b_=row/Nn, n=row%Nn;
    float gir=acc[0][r4]+bih[j], giz=acc[1][r4]+bih[CO+j], gin=acc[2][r4]+bih[2*CO+j];
    float ghr=acc[3][r4]+bhh[j], ghz=acc[4][r4]+bhh[CO+j], ghn=acc[5][r4]+bhh[2*CO+j];
    float r=1.f/(1.f+__expf(-(gir+ghr)));
    float z=1.f/(1.f+__expf(-(giz+ghz)));
    float nn=tanhf(gin + r*ghn);
    float h2=(1.f-z)*nn + z*hs[lr][j];
    h[row*CO+j]=h2;
    rout[((b_*CO+j)*Nn+n)*Tt+t]=h2;
  }
}

// ---------------- K11: STFusion + residual + channel LN -> out ----------
__global__ void k_final(const float* __restrict__ x, const float* __restrict__ tout,
                        const float* __restrict__ rout, const float* __restrict__ alp,
                        const float* __restrict__ rw, const float* __restrict__ rb,
                        const float* __restrict__ lw, const float* __restrict__ lb,
                        float* __restrict__ out){
  int wave=threadIdx.x>>5, lane=threadIdx.x&31;
  int row = blockIdx.x*8 + wave;                // over B*N*T
  int t=row%Tt; int r2=row/Tt; int n=r2%Nn; int b_=r2/Nn;
  float al = alp[0];
  float xv = x[(b_*Nn+n)*Tt + t];
  float vals[2], s=0.f, s2=0.f;
#pragma unroll
  for(int i=0;i<2;i++){
    int c=lane+32*i;
    int idx=((b_*CO+c)*Nn+n)*Tt+t;
    float fused=fmaxf(al*tout[idx]+(1.f-al)*rout[idx],0.f);
    float v=fmaxf(rw[c]*xv+rb[c]+fused,0.f);
    vals[i]=v; s+=v; s2+=v*v;
  }
  s=wsum(s); s2=wsum(s2);
  float mean=s/64.f, var=s2/64.f-mean*mean, rs=rsqrtf(var+1e-5f);
#pragma unroll
  for(int i=0;i<2;i++){
    int c=lane+32*i;
    out[((b_*Nn+n)*CO+c)*Tt+t]=(vals[i]-mean)*rs*lw[c]+lb[c];
  }
}

// ------------------------------- launch ----------------------------------
extern "C" void kernel_launch(void* const* d_in, const int* in_sizes, int n_in,
                              void* d_out, int out_size, void* d_ws, size_t ws_size,
                              hipStream_t stream) {
  const float* x      = (const float*)d_in[0];
  const float* res_att= (const float*)d_in[1];
  const float* pos_T  = (const float*)d_in[2];
  const float* eTw    = (const float*)d_in[3];
  const float* eTb    = (const float*)d_in[4];
  const float* Wq_t   = (const float*)d_in[5];
  const float* Wk_t   = (const float*)d_in[6];
  const float* Wv_t   = (const float*)d_in[7];
  const float* fc_t   = (const float*)d_in[8];
  const float* preW   = (const float*)d_in[9];
  const float* preB   = (const float*)d_in[10];
  const float* pos_S  = (const float*)d_in[11];
  const float* eSw    = (const float*)d_in[12];
  const float* eSb    = (const float*)d_in[13];
  const float* Wq_s   = (const float*)d_in[14];
  const float* Wk_s   = (const float*)d_in[15];
  const float* cpoly  = (const float*)d_in[16];
  const float* adj    = (const float*)d_in[17];
  const float* cmask  = (const float*)d_in[18];
  const float* Theta  = (const float*)d_in[19];
  const float* w3     = (const float*)d_in[20];
  const float* b3     = (const float*)d_in[21];
  const float* w5     = (const float*)d_in[22];
  const float* b5     = (const float*)d_in[23];
  const float* w7     = (const float*)d_in[24];
  const float* b7     = (const float*)d_in[25];
  const float* fW     = (const float*)d_in[26];
  const float* fb     = (const float*)d_in[27];
  const float* Wih    = (const float*)d_in[28];
  const float* Whh    = (const float*)d_in[29];
  const float* bih    = (const float*)d_in[30];
  const float* bhh    = (const float*)d_in[31];
  const float* alpha  = (const float*)d_in[32];
  const float* rw     = (const float*)d_in[33];
  const float* rb     = (const float*)d_in[34];
  const float* lw     = (const float*)d_in[35];
  const float* lb     = (const float*)d_in[36];

  float* out        = (float*)d_out;                       // (B,N,C,T)
  float* scores_out = (float*)d_out + (size_t)Bq*Nn*CO*Tt; // (B,1,H,T,T)

  // workspace carve-up (floats)
  float* ws = (float*)d_ws;
  size_t off=0;
  float* TE   = ws+off; off += (size_t)Bq*Tt*Nn;
  float* qT   = ws+off; off += (size_t)Bq*Tt*96;
  float* kT   = ws+off; off += (size_t)Bq*Tt*96;
  float* vT   = ws+off; off += (size_t)Bq*Tt*96;
  float* ctx  = ws+off; off += (size_t)Bq*Tt*96;
  float* TAT  = ws+off; off += (size_t)Bq*Tt*Nn;
  float* SE   = ws+off; off += (size_t)Bq*Nn*DM;
  float* proj = ws+off; off += (size_t)Bq*Nn*192;
  float* rhs  = ws+off; off += (size_t)Bq*KC*Nn*Tt;
  float* X    = ws+off; off += (size_t)Bq*CO*Nn*Tt;
  float* tout = ws+off; off += (size_t)Bq*CO*Nn*Tt;
  float* rout = ws+off; off += (size_t)Bq*CO*Nn*Tt;
  float* h    = ws+off; off += (size_t)Bq*Nn*CO;

  k_temb  <<<Bq*Tt, 256, 0, stream>>>(x, pos_T, eTw, eTb, TE);
  k_tqkv  <<<Bq*Tt,  96, 0, stream>>>(TE, Wq_t, Wk_t, Wv_t, qT, kT, vT);
  k_tattn <<<Bq*HT, 384, 0, stream>>>(qT, kT, vT, res_att, scores_out, ctx);
  k_tat   <<<Bq*Tt, 256, 0, stream>>>(ctx, fc_t, TE, TAT);
  k_sememb<<<Bq*Nn, 256, 0, stream>>>(TAT, preW, preB, pos_S, eSw, eSb, SE);
  k_proj  <<<(Bq*Nn/16)*12/8, 256, 0, stream>>>(SE, Wq_s, Wk_s, proj);
  k_spat  <<<Bq*KC*45, 256, 0, stream>>>(proj, adj, cmask, cpoly, x, rhs);
  {
    int total = Bq*CO*Nn*Tt;
    k_theta<<<(total+255)/256, 256, 0, stream>>>(rhs, Theta, X);
  }
  k_gtu   <<<Bq*Nn/GTR, 512, 0, stream>>>(X, w3,b3, w5,b5, w7,b7, fW, fb, tout);
  for(int t=0;t<Tt;t++)
    k_gru <<<Bq*Nn/GR, 256, 0, stream>>>(X, Wih, Whh, bih, bhh, h, rout, t);
  k_final <<<Bq*Nn*Tt/8, 256, 0, stream>>>(x, tout, rout, alpha, rw, rb, lw, lb, out);
}